// GIN_27084063769014
// MI455X (gfx1250) — compile-verified
//
#include <hip/hip_runtime.h>

typedef __attribute__((ext_vector_type(2))) float v2f;
typedef __attribute__((ext_vector_type(8))) float v8f;

// -------------------- edge scatter-add (segment_sum) --------------------
// One thread handles one (edge, 4-feature) chunk: coalesced float4 gather of
// x[src] row + 4 global_atomic_add_f32 into aggr[dst] row. Feature tables are
// L2-resident (12.8/25.6 MB << 192 MB L2), so this runs at L2/atomic speed.
template<int F>
__global__ __launch_bounds__(256)
void scatter_add_kernel(const float* __restrict__ x,
                        const int* __restrict__ src,
                        const int* __restrict__ dst,
                        float* __restrict__ aggr,
                        int n_edges) {
    constexpr int PER_EDGE = F / 4;
    long long idx = (long long)blockIdx.x * blockDim.x + threadIdx.x;
    long long total = (long long)n_edges * PER_EDGE;
    if (idx >= total) return;
    int e = (int)(idx / PER_EDGE);
    int c = (int)(idx % PER_EDGE) * 4;
    int s = src[e];
    int d = dst[e];
    const float4 v = *reinterpret_cast<const float4*>(x + (long long)s * F + c);
    float* o = aggr + (long long)d * F + c;
    atomicAdd(o + 0, v.x);
    atomicAdd(o + 1, v.y);
    atomicAdd(o + 2, v.z);
    atomicAdd(o + 3, v.w);
}

// -------------------- fused GIN MLP: relu((x+aggr)@wa+ba)@wb+bb ------------
// One 16-node tile per block; HID/16 waves. Each wave computes one 16x16
// output tile per GEMM via V_WMMA_F32_16X16X4_F32 (fp32 — GEMMs are only
// ~5 GFLOP total, so precision is free and accuracy is preserved).
template<int INF, int HIDF, int OUTF>
__global__ __launch_bounds__((HIDF / 16) * 32)
void gin_mlp_kernel(const float* __restrict__ x,
                    const float* __restrict__ aggr,
                    const float* __restrict__ wa, const float* __restrict__ ba,
                    const float* __restrict__ wb, const float* __restrict__ bb,
                    float* __restrict__ out, int n_nodes) {
    constexpr int WAVES = HIDF / 16;
    __shared__ float lds_in[16][INF + 2];    // padded: kills bank conflicts, keeps 8B align
    __shared__ float lds_h[16][HIDF + 2];

    const int tid   = threadIdx.x;
    const int lane  = tid & 31;
    const int wave  = tid >> 5;
    const int row16 = lane & 15;   // M (A/B column-lane) index
    const int half  = lane >> 4;   // 0: K=k0+0/1  1: K=k0+2/3
    const long long node0 = (long long)blockIdx.x * 16;

    // Stage (1+eps)*x + aggr  (eps = 0) into LDS, coalesced.
    for (int i = tid; i < 16 * INF; i += WAVES * 32) {
        int r = i / INF, c = i % INF;
        long long nr = node0 + r;
        lds_in[r][c] = (nr < n_nodes) ? (x[nr * INF + c] + aggr[nr * INF + c]) : 0.0f;
    }
    __syncthreads();

    // GEMM1: [16 x INF] @ [INF x HIDF] -> relu(+bias) -> LDS
    {
        const int n0 = wave * 16;
        v8f acc = {};
#pragma unroll
        for (int k0 = 0; k0 < INF; k0 += 4) {
            const int ka = k0 + 2 * half;
            v2f a = *reinterpret_cast<const v2f*>(&lds_in[row16][ka]);
            v2f b;
            b.x = wa[(long long)(ka + 0) * HIDF + n0 + row16];
            b.y = wa[(long long)(ka + 1) * HIDF + n0 + row16];
            acc = __builtin_amdgcn_wmma_f32_16x16x4_f32(
                false, a, false, b, (short)0, acc, false, false);
        }
        const float bias = ba[n0 + row16];
#pragma unroll
        for (int v = 0; v < 8; ++v) {
            float h = acc[v] + bias;
            lds_h[v + 8 * half][n0 + row16] = h > 0.0f ? h : 0.0f;
        }
    }
    __syncthreads();

    // GEMM2: [16 x HIDF] @ [HIDF x OUTF] -> +bias -> global
    if (wave < OUTF / 16) {
        const int n0 = wave * 16;
        v8f acc = {};
#pragma unroll
        for (int k0 = 0; k0 < HIDF; k0 += 4) {
            const int ka = k0 + 2 * half;
            v2f a = *reinterpret_cast<const v2f*>(&lds_h[row16][ka]);
            v2f b;
            b.x = wb[(long long)(ka + 0) * OUTF + n0 + row16];
            b.y = wb[(long long)(ka + 1) * OUTF + n0 + row16];
            acc = __builtin_amdgcn_wmma_f32_16x16x4_f32(
                false, a, false, b, (short)0, acc, false, false);
        }
        const float bias = bb[n0 + row16];
#pragma unroll
        for (int v = 0; v < 8; ++v) {
            long long nr = node0 + v + 8 * half;
            if (nr < n_nodes) out[nr * OUTF + n0 + row16] = acc[v] + bias;
        }
    }
}

extern "C" void kernel_launch(void* const* d_in, const int* in_sizes, int n_in,
                              void* d_out, int out_size, void* d_ws, size_t ws_size,
                              hipStream_t stream) {
    const float* x   = (const float*)d_in[0];
    const int*   ei  = (const int*)d_in[1];
    const float* w1a = (const float*)d_in[2];
    const float* b1a = (const float*)d_in[3];
    const float* w1b = (const float*)d_in[4];
    const float* b1b = (const float*)d_in[5];
    const float* w2a = (const float*)d_in[6];
    const float* b2a = (const float*)d_in[7];
    const float* w2b = (const float*)d_in[8];
    const float* b2b = (const float*)d_in[9];
    float* out = (float*)d_out;

    const int N = in_sizes[0] / 64;   // 50000
    const int E = in_sizes[1] / 2;    // 800000
    const int* src = ei;              // edge_index[0]
    const int* dst = ei + E;          // edge_index[1]

    // Workspace layout: aggr1[N*64] | h1[N*128] | aggr2[N*128]  (~64 MB)
    float* aggr1 = (float*)d_ws;
    float* h1    = aggr1 + (size_t)N * 64;
    float* aggr2 = h1 + (size_t)N * 128;

    hipMemsetAsync(aggr1, 0, (size_t)N * 64  * sizeof(float), stream);
    hipMemsetAsync(aggr2, 0, (size_t)N * 128 * sizeof(float), stream);

    // ---- layer 1 ----
    {
        long long total = (long long)E * (64 / 4);
        int blocks = (int)((total + 255) / 256);
        scatter_add_kernel<64><<<blocks, 256, 0, stream>>>(x, src, dst, aggr1, E);
    }
    {
        int blocks = (N + 15) / 16;   // 3125
        gin_mlp_kernel<64, 128, 128><<<blocks, 256, 0, stream>>>(
            x, aggr1, w1a, b1a, w1b, b1b, h1, N);
    }

    // ---- layer 2 ----
    {
        long long total = (long long)E * (128 / 4);
        int blocks = (int)((total + 255) / 256);
        scatter_add_kernel<128><<<blocks, 256, 0, stream>>>(h1, src, dst, aggr2, E);
    }
    {
        int blocks = (N + 15) / 16;
        gin_mlp_kernel<128, 128, 64><<<blocks, 256, 0, stream>>>(
            h1, aggr2, w2a, b2a, w2b, b2b, out, N);
    }
}